// SelfSupervisedPretraining_24919400251999
// MI455X (gfx1250) — compile-verified
//
#include <hip/hip_runtime.h>
#include <math.h>
#include <stdint.h>

typedef __attribute__((ext_vector_type(2))) float v2f;
typedef __attribute__((ext_vector_type(8))) float v8f;

#define WMMA_F32X4(a, b, c) \
    __builtin_amdgcn_wmma_f32_16x16x4_f32(false, (a), false, (b), (short)0, (c), false, false)

__device__ __forceinline__ float gelu_exact(float x) {
    // 0.5 * x * (1 + erf(x / sqrt(2)))  -- matches jax.nn.gelu(approximate=False)
    return 0.5f * x * (1.0f + erff(x * 0.70710678118654752440f));
}

// CDNA5 async copy: global -> LDS, 16 bytes per lane, tracked by ASYNCcnt.
// LDS operand is the byte offset (low 32 bits of the flat shared pointer).
__device__ __forceinline__ void async_gather_b128(void* lds_dst, const void* gsrc) {
    uint32_t lds_off = (uint32_t)(uintptr_t)lds_dst;
    asm volatile("global_load_async_to_lds_b128 %0, %1, off"
                 :: "v"(lds_off), "v"(gsrc)
                 : "memory");
}
__device__ __forceinline__ void async_wait0() {
    asm volatile("s_wait_asynccnt 0x0" ::: "memory");
}

// ---------------------------------------------------------------------------
// Edge branch: gather concat(x[row], x[col]) [E,128] -> Linear(128,256) -> LN
// -> GELU -> Linear(256,1). One wave per 16 edges. Hidden tile (16 n-tiles)
// lives entirely in registers; k-outer loop batches B loads per k-step so the
// WMMA chain pipelines instead of stalling on s_wait_loadcnt 0 per WMMA.
// ---------------------------------------------------------------------------
__global__ __launch_bounds__(128) void edge_mlp_kernel(
    const float* __restrict__ x, const int* __restrict__ ei,
    const float* __restrict__ w1, const float* __restrict__ b1,
    const float* __restrict__ g,  const float* __restrict__ be,
    const float* __restrict__ w2, const float* __restrict__ b2,
    float* __restrict__ out, int N, int E)
{
    __shared__ float lds_a[4][16 * 128];   // per-wave A tile: 16 edges x 128 feats
    const int wave = threadIdx.x >> 5;
    const int lane = threadIdx.x & 31;
    const int hl   = lane >> 4;            // half-wave: 0 or 1
    const int ln16 = lane & 15;
    const long eBase = (long)blockIdx.x * 64 + (long)wave * 16;

    const int* rowIdx = ei;        // edge_index[0][:]
    const int* colIdx = ei + E;    // edge_index[1][:]

    float* A = lds_a[wave];
    // Async gather: lanes 0-15 fetch x[row] (first 64 feats), lanes 16-31
    // fetch x[col] (second 64). One B128 per lane per edge.
    for (int e = 0; e < 16; ++e) {
        long eg = eBase + e; if (eg >= E) eg = E - 1;
        const int idx = (hl == 0) ? rowIdx[eg] : colIdx[eg];
        async_gather_b128(&A[e * 128 + hl * 64 + ln16 * 4],
                          x + (long)idx * 64 + ln16 * 4);
    }
    async_wait0();
    __syncthreads();

    // Accumulators: 16 n-tiles of [16x16] f32 (C layout: M = v + 8*hl, N = ln16)
    v8f h[16];
#pragma unroll
    for (int t = 0; t < 16; ++t) {
        const float bias = b1[t * 16 + ln16];
#pragma unroll
        for (int v = 0; v < 8; ++v) h[t][v] = bias;
    }

    // GEMM1: K = 128 (32 k-steps of 4). Per k-step: 1 LDS b64 for A,
    // 32-load clause for B (rows kb, kb+1 across all 16 n-tiles), 16 WMMAs.
#pragma unroll 2
    for (int k = 0; k < 32; ++k) {
        const int kb = k * 4 + hl * 2;       // A/B: K = 2*(lane/16) + vgpr
        v2f a = *(const v2f*)&A[ln16 * 128 + kb];
        const float* w1r0 = w1 + (long)(kb + 0) * 256 + ln16;
        const float* w1r1 = w1 + (long)(kb + 1) * 256 + ln16;
        float bx[16], by[16];
#pragma unroll
        for (int t = 0; t < 16; ++t) { bx[t] = w1r0[t * 16]; by[t] = w1r1[t * 16]; }
#pragma unroll
        for (int t = 0; t < 16; ++t) {
            v2f b; b.x = bx[t]; b.y = by[t];
            h[t] = WMMA_F32X4(a, b, h[t]);
        }
    }

    // LayerNorm stats per row (row M = v + 8*hl), reduce over 16-lane group
    float mu[8], rstd[8];
#pragma unroll
    for (int v = 0; v < 8; ++v) {
        float p1 = 0.0f, p2 = 0.0f;
#pragma unroll
        for (int t = 0; t < 16; ++t) { float y = h[t][v]; p1 += y; p2 += y * y; }
        for (int d = 1; d < 16; d <<= 1) {
            p1 += __shfl_xor(p1, d, 32);
            p2 += __shfl_xor(p2, d, 32);
        }
        const float m = p1 * (1.0f / 256.0f);
        const float var = p2 * (1.0f / 256.0f) - m * m;
        mu[v] = m; rstd[v] = rsqrtf(var + 1e-5f);
    }

    // Normalize + GELU + dot with w2[256,1]
    float dot[8];
#pragma unroll
    for (int v = 0; v < 8; ++v) dot[v] = 0.0f;
#pragma unroll
    for (int t = 0; t < 16; ++t) {
        const float gg = g[t * 16 + ln16];
        const float bb = be[t * 16 + ln16];
        const float ww = w2[t * 16 + ln16];
#pragma unroll
        for (int v = 0; v < 8; ++v) {
            float y = (h[t][v] - mu[v]) * rstd[v] * gg + bb;
            dot[v] += gelu_exact(y) * ww;
        }
    }
#pragma unroll
    for (int v = 0; v < 8; ++v)
        for (int d = 1; d < 16; d <<= 1) dot[v] += __shfl_xor(dot[v], d, 32);

    if (ln16 == 0) {
        const float b2v = b2[0];
#pragma unroll
        for (int v = 0; v < 8; ++v) {
            long eg = eBase + v + 8 * hl;
            if (eg < E) out[eg] = dot[v] + b2v;
        }
    }
}

// ---------------------------------------------------------------------------
// Node branches: x[N,64] -> Linear(64,H) -> LN(H) -> GELU -> Linear(H,O).
// HT = H/16 tiles, OT = O/16 tiles. One wave per 16 rows, 2 waves per block.
// GELU output staged to LDS row-major to feed WMMA #2 A-operand.
// ---------------------------------------------------------------------------
template <int HT, int OT>
__global__ __launch_bounds__(64) void node_mlp_kernel(
    const float* __restrict__ x,
    const float* __restrict__ w1, const float* __restrict__ b1,
    const float* __restrict__ g,  const float* __restrict__ be,
    const float* __restrict__ w2, const float* __restrict__ b2,
    float* __restrict__ out, int N)
{
    constexpr int H = HT * 16;
    constexpr int O = OT * 16;
    __shared__ float xs[2][16 * 64];
    __shared__ float hs[2][16 * H];

    const int wave = threadIdx.x >> 5;
    const int lane = threadIdx.x & 31;
    const int hl   = lane >> 4;
    const int ln16 = lane & 15;
    const long rBase = (long)blockIdx.x * 32 + (long)wave * 16;

    float* X = xs[wave];
    // Async load of 16 rows x 64 floats: per iter, half-wave 0 takes row 2e,
    // half-wave 1 takes row 2e+1; 16 lanes x B128 = one 64-float row.
    for (int e = 0; e < 8; ++e) {
        const int rr = e * 2 + hl;
        long r = rBase + rr; if (r >= N) r = N - 1;
        async_gather_b128(&X[rr * 64 + ln16 * 4], x + r * 64 + ln16 * 4);
    }
    async_wait0();
    __syncthreads();

    // GEMM1: [16,64] x [64,H], k-outer with batched B loads
    v8f h[HT];
#pragma unroll
    for (int t = 0; t < HT; ++t) {
        const float bias = b1[t * 16 + ln16];
#pragma unroll
        for (int v = 0; v < 8; ++v) h[t][v] = bias;
    }
#pragma unroll 2
    for (int k = 0; k < 16; ++k) {
        const int kb = k * 4 + hl * 2;
        v2f a = *(const v2f*)&X[ln16 * 64 + kb];
        const float* w1r0 = w1 + (long)(kb + 0) * H + ln16;
        const float* w1r1 = w1 + (long)(kb + 1) * H + ln16;
        float bx[HT], by[HT];
#pragma unroll
        for (int t = 0; t < HT; ++t) { bx[t] = w1r0[t * 16]; by[t] = w1r1[t * 16]; }
#pragma unroll
        for (int t = 0; t < HT; ++t) {
            v2f b; b.x = bx[t]; b.y = by[t];
            h[t] = WMMA_F32X4(a, b, h[t]);
        }
    }

    // LayerNorm over H
    float mu[8], rstd[8];
#pragma unroll
    for (int v = 0; v < 8; ++v) {
        float p1 = 0.0f, p2 = 0.0f;
#pragma unroll
        for (int t = 0; t < HT; ++t) { float y = h[t][v]; p1 += y; p2 += y * y; }
        for (int d = 1; d < 16; d <<= 1) {
            p1 += __shfl_xor(p1, d, 32);
            p2 += __shfl_xor(p2, d, 32);
        }
        const float m = p1 * (1.0f / (float)H);
        const float var = p2 * (1.0f / (float)H) - m * m;
        mu[v] = m; rstd[v] = rsqrtf(var + 1e-5f);
    }

    // Normalize + GELU -> LDS (row-major [16][H]) for GEMM2 A-operand
    float* Hb = hs[wave];
#pragma unroll
    for (int t = 0; t < HT; ++t) {
        const float gg = g[t * 16 + ln16];
        const float bb = be[t * 16 + ln16];
#pragma unroll
        for (int v = 0; v < 8; ++v) {
            const int m = v + 8 * hl;
            float y = (h[t][v] - mu[v]) * rstd[v] * gg + bb;
            Hb[m * H + t * 16 + ln16] = gelu_exact(y);
        }
    }
    __syncthreads();

    // GEMM2: [16,H] x [H,O], k-outer with batched B loads
    v8f acc[OT];
#pragma unroll
    for (int t = 0; t < OT; ++t) {
        const float bias = b2[t * 16 + ln16];
#pragma unroll
        for (int v = 0; v < 8; ++v) acc[t][v] = bias;
    }
#pragma unroll 2
    for (int k = 0; k < H / 4; ++k) {
        const int kb = k * 4 + hl * 2;
        v2f a = *(const v2f*)&Hb[ln16 * H + kb];
        const float* w2r0 = w2 + (long)(kb + 0) * O + ln16;
        const float* w2r1 = w2 + (long)(kb + 1) * O + ln16;
        float bx[OT], by[OT];
#pragma unroll
        for (int t = 0; t < OT; ++t) { bx[t] = w2r0[t * 16]; by[t] = w2r1[t * 16]; }
#pragma unroll
        for (int t = 0; t < OT; ++t) {
            v2f b; b.x = bx[t]; b.y = by[t];
            acc[t] = WMMA_F32X4(a, b, acc[t]);
        }
    }
#pragma unroll
    for (int t = 0; t < OT; ++t) {
#pragma unroll
        for (int v = 0; v < 8; ++v) {
            const long r = rBase + v + 8 * hl;
            if (r < N) out[r * O + t * 16 + ln16] = acc[t][v];
        }
    }
}

extern "C" void kernel_launch(void* const* d_in, const int* in_sizes, int n_in,
                              void* d_out, int out_size, void* d_ws, size_t ws_size,
                              hipStream_t stream) {
    (void)n_in; (void)out_size; (void)d_ws; (void)ws_size;

    const float* x     = (const float*)d_in[0];
    const int*   ei    = (const int*)  d_in[1];
    const float* sp_w1 = (const float*)d_in[2];
    const float* sp_b1 = (const float*)d_in[3];
    const float* sp_g  = (const float*)d_in[4];
    const float* sp_be = (const float*)d_in[5];
    const float* sp_w2 = (const float*)d_in[6];
    const float* sp_b2 = (const float*)d_in[7];
    const float* fr_w1 = (const float*)d_in[8];
    const float* fr_b1 = (const float*)d_in[9];
    const float* fr_g  = (const float*)d_in[10];
    const float* fr_be = (const float*)d_in[11];
    const float* fr_w2 = (const float*)d_in[12];
    const float* fr_b2 = (const float*)d_in[13];
    const float* ph_w1 = (const float*)d_in[14];
    const float* ph_b1 = (const float*)d_in[15];
    const float* ph_g  = (const float*)d_in[16];
    const float* ph_be = (const float*)d_in[17];
    const float* ph_w2 = (const float*)d_in[18];
    const float* ph_b2 = (const float*)d_in[19];

    const int N = in_sizes[0] / 64;   // 50000
    const int E = in_sizes[1] / 2;    // 800000

    float* out_sp = (float*)d_out;                       // [E,1]
    float* out_fr = out_sp + (long)E;                    // [N,64]
    float* out_ph = out_fr + (long)N * 64;               // [N,128]

    // Edge branch: 64 edges per 128-thread block (4 waves x 16 edges)
    const int eBlocks = (E + 63) / 64;
    edge_mlp_kernel<<<eBlocks, 128, 0, stream>>>(
        x, ei, sp_w1, sp_b1, sp_g, sp_be, sp_w2, sp_b2, out_sp, N, E);

    // Node branches: 32 rows per 64-thread block (2 waves x 16 rows)
    const int nBlocks = (N + 31) / 32;
    node_mlp_kernel<8, 4><<<nBlocks, 64, 0, stream>>>(
        x, fr_w1, fr_b1, fr_g, fr_be, fr_w2, fr_b2, out_fr, N);
    node_mlp_kernel<16, 8><<<nBlocks, 64, 0, stream>>>(
        x, ph_w1, ph_b1, ph_g, ph_be, ph_w2, ph_b2, out_ph, N);
}